// PIKF_layer_18081812316440
// MI455X (gfx1250) — compile-verified
//
#include <hip/hip_runtime.h>

typedef __attribute__((ext_vector_type(2))) float v2f;
typedef __attribute__((ext_vector_type(8))) float v8f;

#define WAVES_PER_BLOCK 8
#define CHUNK 8   // M-tiles swept per wave (A operand reused across them)

#define INV2PI 0.15915494309189535

// Asymptotic branch of J0(x)/(2pi) for x (clamped to >= 8 internally).
// 1/(2pi) folded into the amplitude constant. Hardware v_rcp/v_rsq/v_sin/v_cos
// with explicit revolution-range reduction (hw sin valid to +-256 revolutions;
// k*r reaches ~1600 revolutions here).
__device__ __forceinline__ float j0_large_i2p(float x) {
  float axb = __builtin_fmaxf(x, 8.0f);
  float z  = 8.0f * __builtin_amdgcn_rcpf(axb);
  float y2 = z * z;
  float xx = axb - 0.785398164f;  // ax - pi/4
  float p0 = __builtin_fmaf(y2,
             __builtin_fmaf(y2,
             __builtin_fmaf(y2,
             __builtin_fmaf(y2, 0.2093887211e-6f, -0.2073370639e-5f),
                            0.2734510407e-4f), -0.1098628627e-2f), 1.0f);
  float q0 = __builtin_fmaf(y2,
             __builtin_fmaf(y2,
             __builtin_fmaf(y2,
             __builtin_fmaf(y2, -0.934935152e-7f, 0.7621095161e-6f),
                            -0.6911147651e-5f), 0.1430488765e-3f),
                            -0.1562499995e-1f);
  float f = xx * (float)INV2PI;   // revolutions
  f = f - __builtin_rintf(f);     // [-0.5, 0.5]
  float sn = __builtin_amdgcn_sinf(f);
  float cs = __builtin_amdgcn_cosf(f);
  // sqrt(0.636619772/ax)/(2pi) = (sqrt(2/pi)/(2pi)) * rsqrt(ax)
  float amp = (float)(0.79788456080286536 * INV2PI) * __builtin_amdgcn_rsqf(axb);
  return amp * (cs * p0 - z * sn * q0);
}

// Rational-polynomial branch of J0(x)/(2pi) for x (clamped to <= 8 internally).
// 1/(2pi) folded into the numerator coefficients (compile-time constants).
__device__ __forceinline__ float j0_small_i2p(float x) {
  float axs = __builtin_fminf(x, 8.0f);
  float y = axs * axs;
  float num = __builtin_fmaf(y,
              __builtin_fmaf(y,
              __builtin_fmaf(y,
              __builtin_fmaf(y,
              __builtin_fmaf(y, (float)(-184.9052456 * INV2PI),
                                (float)(77392.33017 * INV2PI)),
                             (float)(-11214424.18 * INV2PI)),
                             (float)(651619640.7 * INV2PI)),
                             (float)(-13362590354.0 * INV2PI)),
                             (float)(57568490574.0 * INV2PI));
  float den = __builtin_fmaf(y,
              __builtin_fmaf(y,
              __builtin_fmaf(y,
              __builtin_fmaf(y,
              __builtin_fmaf(y, 1.0f, 267.8532712f),
                             59272.64853f), 9494680.718f),
                             1029532985.0f), 57568490411.0f);
  return num * __builtin_amdgcn_rcpf(den);
}

__global__ __launch_bounds__(256, 2)
void pikf_j0_wmma_kernel(const float* __restrict__ p,     // (N,2)
                         const float* __restrict__ s,     // (M,2)
                         float* __restrict__ out,         // (N,M)
                         int N, int M, int tilesM, int chunks) {
  const int lane = threadIdx.x & 31;
  const int wv   = threadIdx.x >> 5;
  const long gw  = (long)blockIdx.x * WAVES_PER_BLOCK + wv;

  const int tilesN = (N + 15) >> 4;
  const int tileN  = (int)(gw / chunks);
  if (tileN >= tilesN) return;              // wave-uniform: EXEC stays all-1s
  const int chunk  = (int)(gw - (long)tileN * chunks);

  const bool hi = lane >= 16;
  const int  m  = lane & 15;

  // Edge tiles shifted back to fully overlap the valid range: overlapping
  // waves write byte-identical values (benign, deterministic race).
  const int n0 = min(tileN * 16, N - 16);

  // ---- A operand: 16x4 fp32, rows = [k*px, k*py, k^2*|p|^2, 1] ----
  // (k folded into the Gram operands so the WMMA yields (k*r)^2 directly)
  // VGPR0 = K0 (lanes 0-15) / K2 (lanes 16-31); VGPR1 = K1 / K3
  const int row = n0 + m;
  float px = p[2 * row + 0];
  float py = p[2 * row + 1];
  float np = __builtin_fmaf(px, px, py * py);
  v2f A;
  A.x = hi ? 1.0e6f * np : 1000.0f * px;    // K0 | K2
  A.y = hi ? 1.0f        : 1000.0f * py;    // K1 | K3

  const int tmBeg = chunk * CHUNK;
  const int tmEnd = min(tmBeg + CHUNK, tilesM);
  const size_t rowBase = (size_t)(n0 + (hi ? 8 : 0)) * (size_t)M;

  for (int tm = tmBeg; tm < tmEnd; ++tm) {
    // ---- B operand: 4x16 fp32, cols = [-2k*sx, -2k*sy, 1, k^2*|s|^2] ----
    const int c0  = min(tm * 16, M - 16);   // overlap-shifted edge tile
    const int col = c0 + m;
    float sx = s[2 * col + 0];
    float sy = s[2 * col + 1];
    float ns = __builtin_fmaf(sx, sx, sy * sy);
    v2f B;
    B.x = hi ? 1.0f        : -2000.0f * sx; // K0 | K2
    B.y = hi ? 1.0e6f * ns : -2000.0f * sy; // K1 | K3

    // D = A x B + 0  ->  16x16 tile of (k*r)^2, exact fp32 FMA math
    v8f acc = {};
    acc = __builtin_amdgcn_wmma_f32_16x16x4_f32(
        /*neg_a=*/false, A, /*neg_b=*/false, B,
        /*c_mod=*/(short)0, acc, /*reuse_a=*/false, /*reuse_b=*/false);

    // x = k*r via single v_med3 clamp + raw v_sqrt
    float x[8];
#pragma unroll
    for (int v = 0; v < 8; ++v) {
      float r2 = __builtin_amdgcn_fmed3f(acc[v], 0.0f, 3.402823466e38f);
      x[v] = __builtin_amdgcn_sqrtf(r2);
    }

    // Asymptotic branch always (dominant path)
    float res[8];
#pragma unroll
    for (int v = 0; v < 8; ++v) res[v] = j0_large_i2p(x[v]);

    // Rational branch only if any lane of the wave needs it (k*r < 8 means
    // r < 8mm here -- essentially never). min-tree + ballot gives a
    // wave-uniform scalar branch, so EXEC stays all-1s either way.
    float mn01 = __builtin_fminf(x[0], x[1]);
    float mn23 = __builtin_fminf(x[2], x[3]);
    float mn45 = __builtin_fminf(x[4], x[5]);
    float mn67 = __builtin_fminf(x[6], x[7]);
    float mn = __builtin_fminf(__builtin_fminf(mn01, mn23),
                               __builtin_fminf(mn45, mn67));
    if (__builtin_amdgcn_ballot_w32(mn < 8.0f) != 0u) {
#pragma unroll
      for (int v = 0; v < 8; ++v) {
        float js = j0_small_i2p(x[v]);
        res[v] = (x[v] < 8.0f) ? js : res[v];
      }
    }

    // 8 unconditional streaming stores (write-once output, > L2)
    float* base = out + rowBase + (size_t)col;
#pragma unroll
    for (int v = 0; v < 8; ++v) {
      __builtin_nontemporal_store(res[v], base + (size_t)v * (size_t)M);
    }
  }
}

// Degenerate fallback (N<16 or M<16): plain per-element evaluation.
__global__ void pikf_j0_small_kernel(const float* __restrict__ p,
                                     const float* __restrict__ s,
                                     float* __restrict__ out, int N, int M) {
  long idx = (long)blockIdx.x * blockDim.x + threadIdx.x;
  if (idx >= (long)N * M) return;
  int i = (int)(idx / M);
  int j = (int)(idx - (long)i * M);
  float dx = p[2 * i + 0] - s[2 * j + 0];
  float dy = p[2 * i + 1] - s[2 * j + 1];
  float x = 1000.0f * __builtin_amdgcn_sqrtf(__builtin_fmaf(dx, dx, dy * dy));
  float jl = j0_large_i2p(x);
  float js = j0_small_i2p(x);
  out[idx] = (x < 8.0f) ? js : jl;
}

extern "C" void kernel_launch(void* const* d_in, const int* in_sizes, int n_in,
                              void* d_out, int out_size, void* d_ws, size_t ws_size,
                              hipStream_t stream) {
  const float* p = (const float*)d_in[0];   // (N,2) float32
  const float* s = (const float*)d_in[1];   // (M,2) float32
  float* out = (float*)d_out;               // (N,M) float32

  const int N = in_sizes[0] / 2;
  const int M = in_sizes[1] / 2;

  if (N >= 16 && M >= 16) {
    const int tilesN = (N + 15) / 16;
    const int tilesM = (M + 15) / 16;
    const int chunks = (tilesM + CHUNK - 1) / CHUNK;
    const long waves = (long)tilesN * chunks;
    const int blocks = (int)((waves + WAVES_PER_BLOCK - 1) / WAVES_PER_BLOCK);
    pikf_j0_wmma_kernel<<<blocks, 32 * WAVES_PER_BLOCK, 0, stream>>>(
        p, s, out, N, M, tilesM, chunks);
  } else {
    const long total = (long)N * M;
    const int blocks = (int)((total + 255) / 256);
    pikf_j0_small_kernel<<<blocks, 256, 0, stream>>>(p, s, out, N, M);
  }
}